// FocusedLinearAttention_35442070127207
// MI455X (gfx1250) — compile-verified
//
#include <hip/hip_runtime.h>
#include <hip/hip_bf16.h>

// Focused Linear Attention on MI455X (gfx1250, wave32).
// fp32 throughout via V_WMMA_F32_16X16X4_F32 (matches reference precision).
// Per-head tiles staged with the Tensor Data Mover (tensor_load_to_lds);
// big GEMMs use double-buffered LDS + 32x32-per-wave register tiles so the
// WMMA stream, not LDS bandwidth, is the limiter.
//
// Shapes: B=4, N=4096, C=1024, H=16, d=64. qkv layout: (B,N,3,H,d) i.e.
// row m = b*N+n has 3C=3072 floats: [q(1024) | k(1024) | v(1024)], head h at h*64.

typedef __attribute__((ext_vector_type(2))) float v2f;
typedef __attribute__((ext_vector_type(8))) float v8f;

#define SEQ_N 4096
#define CH    1024
#define C3    3072
#define EPSF  1e-6f

__device__ __forceinline__ v8f wmma4(v2f a, v2f b, v8f c) {
    // D = A(16x4,f32) * B(4x16,f32) + C(16x16,f32)
    return __builtin_amdgcn_wmma_f32_16x16x4_f32(
        /*neg_a=*/false, a, /*neg_b=*/false, b,
        /*c_mod=*/(short)0, c, /*reuse_a=*/false, /*reuse_b=*/false);
}

// ---------------------------------------------------------------------------
// Tensor Data Mover: 2D tile -> LDS with hardware row padding.
// D# bit layout per CDNA5 ISA ch.8 (§8.3 group0, §8.4 group1).
// ---------------------------------------------------------------------------
#if defined(__has_builtin)
#if __has_builtin(__builtin_amdgcn_tensor_load_to_lds) && \
    __has_builtin(__builtin_amdgcn_s_wait_tensorcnt)
#define FLA_HAVE_TDM 1
#endif
#endif

#ifdef FLA_HAVE_TDM
typedef unsigned int u32;
typedef __attribute__((ext_vector_type(4))) unsigned int v4u;
typedef __attribute__((ext_vector_type(4))) int v4i;
typedef __attribute__((ext_vector_type(8))) int v8i;

// Loads a tile_d0 x tile_d1 tile of f32 (row stride stride_d0 elements) into
// LDS, inserting 4 pad DWORDs after every 64 DWORDs (-> LDS row stride 68).
__device__ __forceinline__ void tdm_load_tile_f32(const float* gptr, const float* lptr,
                                                  u32 tile_d0, u32 tile_d1,
                                                  u32 stride_d0) {
    const u32 lds = (u32)(size_t)lptr;                    // LDS byte offset
    const unsigned long long ga = (unsigned long long)(size_t)gptr;
    v4u g0;
    g0.x = 1u;                                            // count=1 (valid, user mode)
    g0.y = lds;                                           // lds_addr
    g0.z = (u32)(ga & 0xffffffffu);                       // global_addr[31:0]
    g0.w = (u32)((ga >> 32) & 0x01ffffffu) | (2u << 30);  // addr[56:32] | type=2
    v8i g1;
    g1[0] = (int)((2u << 16)      // data_size = 4B
                | (1u << 20)      // pad_enable
                | (5u << 22)      // pad_interval: 64 DWORDs
                | (3u << 25));    // pad_amount:  4 DWORDs
    g1[1] = (int)((tile_d0 /*tensor_dim0*/ & 0xffffu) << 16);
    g1[2] = (int)((tile_d1 /*tensor_dim1*/ & 0xffffu) << 16);
    g1[3] = (int)((tile_d0 & 0xffffu) << 16);             // tile_dim0
    g1[4] = (int)(tile_d1 & 0xffffu);                     // tile_dim1 (tile_dim2 = 0)
    g1[5] = (int)stride_d0;                               // tensor_dim0_stride lo32
    g1[6] = 0;
    g1[7] = 0;
    const v4i z4 = {0, 0, 0, 0};
#if __clang_major__ >= 23
    const v8i z8 = {0, 0, 0, 0, 0, 0, 0, 0};
    __builtin_amdgcn_tensor_load_to_lds(g0, g1, z4, z4, z8, 0);
#else
    __builtin_amdgcn_tensor_load_to_lds(g0, g1, z4, z4, 0);
#endif
}
#endif // FLA_HAVE_TDM

// ---------------------------------------------------------------------------
// Kernel 1: qkv = x @ w_qkv^T      A:(16384,1024) W:(3072,1024) -> (16384,3072)
// WG tile 128x64, BK=16, double-buffered LDS; waves 4x2, 32x32 per wave
// (4 accumulators -> 16 WMMAs per chunk per wave, 1 barrier per chunk).
// ---------------------------------------------------------------------------
__global__ __launch_bounds__(256)
void fla_qkv_gemm(const float* __restrict__ A, const float* __restrict__ W,
                  float* __restrict__ Out) {
    const int tid  = threadIdx.x;
    const int lane = tid & 31;
    const int wid  = tid >> 5;
    const int wm   = wid & 3;          // 4 row groups of 32
    const int wn   = wid >> 2;         // 2 col groups of 32
    const int blockM = blockIdx.x * 128;
    const int blockO = blockIdx.y * 64;

    __shared__ float As[2][128][20];   // 80B row stride: 16B-aligned, conflict-free
    __shared__ float Bs[2][64][20];

    v8f acc00 = {}, acc01 = {}, acc10 = {}, acc11 = {};

    const int lrA = tid >> 1;          // A staging row 0..127
    const int lcA = (tid & 1) * 8;     // A staging col 0 or 8
    const int lrB = tid >> 2;          // B staging row 0..63
    const int lcB = (tid & 3) * 4;     // B staging col 0,4,8,12

    const float* arow = A + (size_t)(blockM + lrA) * CH;
    const float* wrow = W + (size_t)(blockO + lrB) * CH;

    auto stage = [&](int buf, int k0) {
        float4 a0 = *(const float4*)(arow + k0 + lcA);
        float4 a1 = *(const float4*)(arow + k0 + lcA + 4);
        float4 wv = *(const float4*)(wrow + k0 + lcB);
        *(float4*)&As[buf][lrA][lcA]     = a0;
        *(float4*)&As[buf][lrA][lcA + 4] = a1;
        *(float4*)&Bs[buf][lrB][lcB]     = wv;
        __builtin_prefetch(arow + k0 + 16 + lcA, 0, 1);
    };

    stage(0, 0);
    __syncthreads();

    int buf = 0;
    for (int k0 = 0; k0 < CH; k0 += 16) {
        if (k0 + 16 < CH) stage(buf ^ 1, k0 + 16);  // overlap with WMMAs below

        const int kb = (lane >> 4) * 2;   // lanes 16-31 hold K+2,K+3
        const int ln = lane & 15;
        const int rA = wm * 32;
        const int cB = wn * 32;
#pragma unroll
        for (int kk = 0; kk < 16; kk += 4) {
            v2f a0; a0.x = As[buf][rA + ln][kk + kb];
                    a0.y = As[buf][rA + ln][kk + kb + 1];
            v2f a1; a1.x = As[buf][rA + 16 + ln][kk + kb];
                    a1.y = As[buf][rA + 16 + ln][kk + kb + 1];
            v2f b0; b0.x = Bs[buf][cB + ln][kk + kb];
                    b0.y = Bs[buf][cB + ln][kk + kb + 1];
            v2f b1; b1.x = Bs[buf][cB + 16 + ln][kk + kb];
                    b1.y = Bs[buf][cB + 16 + ln][kk + kb + 1];
            acc00 = wmma4(a0, b0, acc00);
            acc01 = wmma4(a0, b1, acc01);
            acc10 = wmma4(a1, b0, acc10);
            acc11 = wmma4(a1, b1, acc11);
        }
        __syncthreads();
        buf ^= 1;
    }

    const int msel = (lane >> 4) * 8;     // C/D: VGPR r -> M = r (+8 for hi lanes)
    const int ln   = lane & 15;
#pragma unroll
    for (int r = 0; r < 8; ++r) {
        const size_t row0 = (size_t)(blockM + wm * 32 + r + msel);
        const size_t row1 = row0 + 16;
        const int    col  = blockO + wn * 32 + ln;
        Out[row0 * C3 + col]      = acc00[r];
        Out[row0 * C3 + col + 16] = acc01[r];
        Out[row1 * C3 + col]      = acc10[r];
        Out[row1 * C3 + col + 16] = acc11[r];
    }
}

// ---------------------------------------------------------------------------
// Kernel 2: focused feature map on q and k rows (in place).
// One wave per (m,h,t) row of d=64: relu+eps -> l1 -> ^3 -> l1.
// ---------------------------------------------------------------------------
__global__ __launch_bounds__(256)
void fla_featmap(float* __restrict__ qkv) {
    const int  wid  = threadIdx.x >> 5;
    const int  lane = threadIdx.x & 31;
    const long rowid = (long)blockIdx.x * 8 + wid;  // B*N*H*2 rows
    const int  t = (int)(rowid & 1);                // 0=q, 1=k
    const long tmp = rowid >> 1;
    const int  h = (int)(tmp & 15);
    const long m = tmp >> 4;

    float* p = qkv + m * C3 + t * CH + h * 64;
    float v0 = p[lane], v1 = p[lane + 32];
    float r0 = fmaxf(v0, 0.f) + EPSF;
    float r1 = fmaxf(v1, 0.f) + EPSF;
    float s = r0 + r1;
#pragma unroll
    for (int off = 16; off; off >>= 1) s += __shfl_xor(s, off, 32);
    r0 /= s; r1 /= s;
    float c0 = r0 * r0 * r0, c1 = r1 * r1 * r1;
    float s2 = c0 + c1;
#pragma unroll
    for (int off = 16; off; off >>= 1) s2 += __shfl_xor(s2, off, 32);
    p[lane]      = c0 / s2;
    p[lane + 32] = c1 / s2;
}

// ---------------------------------------------------------------------------
// Kernel 3: ksum[b,h,e] = sum_n k[b,h,n,e]   (grid = B*H blocks)
// ---------------------------------------------------------------------------
__global__ __launch_bounds__(256)
void fla_ksum(const float* __restrict__ qkv, float* __restrict__ ksum) {
    const int bh = blockIdx.x;
    const int b = bh >> 4, h = bh & 15;
    const int e    = threadIdx.x & 63;
    const int part = threadIdx.x >> 6;   // 0..3

    const float* base = qkv + (size_t)b * SEQ_N * C3 + CH + h * 64 + e;
    float s = 0.f;
    for (int n = part * 1024; n < part * 1024 + 1024; ++n)
        s += base[(size_t)n * C3];

    __shared__ float red[4][64];
    red[part][e] = s;
    __syncthreads();
    if (part == 0)
        ksum[bh * 64 + e] = red[0][e] + red[1][e] + red[2][e] + red[3][e];
}

// ---------------------------------------------------------------------------
// Kernel 4: kv[b,h] = k^T @ v  (64x64, K=4096)   grid = B*H blocks.
// k/v chunks staged to LDS by the TDM (hardware-padded rows).
// ---------------------------------------------------------------------------
__global__ __launch_bounds__(256)
void fla_kv(const float* __restrict__ qkv, float* __restrict__ kv) {
    const int bh = blockIdx.x;
    const int b = bh >> 4, h = bh & 15;
    const int tid = threadIdx.x, lane = tid & 31, wid = tid >> 5;
    const int tm = wid & 3;      // dd tile (16 rows)
    const int tc = wid >> 2;     // e col group (32 cols)

    __shared__ float ks[64][68];
    __shared__ float vs[64][68];
    v8f acc0 = {}, acc1 = {};

    const size_t headK = (size_t)b * SEQ_N * C3 + CH + h * 64;
    const size_t headV = headK + CH;

    for (int n0 = 0; n0 < SEQ_N; n0 += 64) {
#ifdef FLA_HAVE_TDM
        if (wid == 0) {
            tdm_load_tile_f32(qkv + headK + (size_t)n0 * C3, &ks[0][0], 64, 64, C3);
            tdm_load_tile_f32(qkv + headV + (size_t)n0 * C3, &vs[0][0], 64, 64, C3);
            __builtin_amdgcn_s_wait_tensorcnt(0);
        }
#else
        for (int i = 0; i < 16; ++i) {
            const int idx = tid + i * 256;
            const int r = idx >> 6, c = idx & 63;
            ks[r][c] = qkv[headK + (size_t)(n0 + r) * C3 + c];
            vs[r][c] = qkv[headV + (size_t)(n0 + r) * C3 + c];
        }
#endif
        __syncthreads();

        const int kb = (lane >> 4) * 2;
        const int ln = lane & 15;
#pragma unroll
        for (int kk = 0; kk < 64; kk += 4) {
            // A[m=dd][k=n] = ks[n][dd]  (transposed read out of LDS)
            v2f a;  a.x  = ks[kk + kb][tm * 16 + ln];
                    a.y  = ks[kk + kb + 1][tm * 16 + ln];
            v2f b0; b0.x = vs[kk + kb][tc * 32 + ln];
                    b0.y = vs[kk + kb + 1][tc * 32 + ln];
            v2f b1; b1.x = vs[kk + kb][tc * 32 + 16 + ln];
                    b1.y = vs[kk + kb + 1][tc * 32 + 16 + ln];
            acc0 = wmma4(a, b0, acc0);
            acc1 = wmma4(a, b1, acc1);
        }
        __syncthreads();
    }

    const int msel = (lane >> 4) * 8;
    const int ln   = lane & 15;
    float* out = kv + (size_t)bh * 4096;
#pragma unroll
    for (int r = 0; r < 8; ++r) {
        const int dd = tm * 16 + r + msel;
        out[dd * 64 + tc * 32 + ln]      = acc0[r];
        out[dd * 64 + tc * 32 + 16 + ln] = acc1[r];
    }
}

// ---------------------------------------------------------------------------
// Kernel 5: out[n,e] = z_n * (q[n,:] @ kv)   written in place over q region.
// grid = (B*H, N/128); q tile + kv staged by TDM.
// ---------------------------------------------------------------------------
__global__ __launch_bounds__(256)
void fla_apply(float* __restrict__ qkv, const float* __restrict__ kv,
               const float* __restrict__ ksum) {
    const int bh = blockIdx.x;
    const int b = bh >> 4, h = bh & 15;
    const int n0 = blockIdx.y * 128;
    const int tid = threadIdx.x, lane = tid & 31, wid = tid >> 5;

    __shared__ float qs[128][68];
    __shared__ float kvs[64][68];
    __shared__ float zs[128];

    const size_t headQ = ((size_t)b * SEQ_N + n0) * C3 + h * 64;

#ifdef FLA_HAVE_TDM
    if (wid == 0) {
        tdm_load_tile_f32(qkv + headQ, &qs[0][0], 64, 128, C3);
        tdm_load_tile_f32(kv + (size_t)bh * 4096, &kvs[0][0], 64, 64, 64);
        __builtin_amdgcn_s_wait_tensorcnt(0);
    }
#else
    for (int i = 0; i < 32; ++i) {
        const int idx = tid + i * 256;
        const int r = idx >> 6, c = idx & 63;
        qs[r][c] = qkv[headQ + (size_t)r * C3 + c];
    }
    for (int i = 0; i < 16; ++i) {
        const int idx = tid + i * 256;
        kvs[idx >> 6][idx & 63] = kv[(size_t)bh * 4096 + idx];
    }
#endif
    __syncthreads();

    if (tid < 128) {
        const float* kp = ksum + bh * 64;
        float dot = 0.f;
#pragma unroll 8
        for (int e = 0; e < 64; ++e) dot += qs[tid][e] * kp[e];
        zs[tid] = 1.0f / (dot + EPSF);
    }
    __syncthreads();

    v8f acc[4] = {{}, {}, {}, {}};
    const int kb = (lane >> 4) * 2;
    const int ln = lane & 15;
    const int rbase = wid * 16;       // 8 waves x 16 rows = 128
#pragma unroll
    for (int kk = 0; kk < 64; kk += 4) {
        v2f a; a.x = qs[rbase + ln][kk + kb];
               a.y = qs[rbase + ln][kk + kb + 1];
#pragma unroll
        for (int j = 0; j < 4; ++j) {
            v2f bb; bb.x = kvs[kk + kb][j * 16 + ln];
                    bb.y = kvs[kk + kb + 1][j * 16 + ln];
            acc[j] = wmma4(a, bb, acc[j]);
        }
    }

    const int msel = (lane >> 4) * 8;
#pragma unroll
    for (int j = 0; j < 4; ++j)
#pragma unroll
        for (int r = 0; r < 8; ++r) {
            const int row = rbase + r + msel;
            qkv[headQ + (size_t)row * C3 + j * 16 + ln] = acc[j][r] * zs[row];
        }
}

// ---------------------------------------------------------------------------
// Kernel 6: final projection with fused depthwise-conv3 + residual + bias.
// attn rows live in the q region of qkv (row stride 3C). Double-buffered LDS,
// WG tile 128x64, 32x32 per wave.
// d_out[m,o] = sum_c (attn[m,c] + conv3(attn)[m,c]) * w_proj[o,c] + b_proj[o]
// ---------------------------------------------------------------------------
__global__ __launch_bounds__(256)
void fla_proj(const float* __restrict__ attn, const float* __restrict__ w_proj,
              const float* __restrict__ b_proj, const float* __restrict__ w_dwc,
              const float* __restrict__ b_dwc, float* __restrict__ out) {
    const int tid = threadIdx.x, lane = tid & 31, wid = tid >> 5;
    const int wm = wid & 3, wn = wid >> 2;
    const int blockM = blockIdx.x * 128;
    const int blockO = blockIdx.y * 64;

    __shared__ float As[2][128][20];
    __shared__ float Bs[2][64][20];
    v8f acc00 = {}, acc01 = {}, acc10 = {}, acc11 = {};

    const int lrA = tid >> 1;          // 0..127
    const int lcA = (tid & 1) * 8;     // 0 or 8
    const int lrB = tid >> 2;          // 0..63
    const int lcB = (tid & 3) * 4;

    const int m  = blockM + lrA;
    const int n  = m & (SEQ_N - 1);
    const float* rowC = attn + (size_t)m * C3;
    const float* rowP = attn + (size_t)(m - 1) * C3;
    const float* rowN = attn + (size_t)(m + 1) * C3;
    const bool hasP = (n > 0);
    const bool hasN = (n < SEQ_N - 1);
    const float* wrow = w_proj + (size_t)(blockO + lrB) * CH;

    auto stage = [&](int buf, int k0) {
#pragma unroll
        for (int j = 0; j < 8; ++j) {
            const int c = k0 + lcA + j;
            const float ctr = rowC[c];
            const float a0  = hasP ? rowP[c] : 0.f;
            const float a2  = hasN ? rowN[c] : 0.f;
            const float cv  = a0 * w_dwc[c * 3] + ctr * w_dwc[c * 3 + 1] +
                              a2 * w_dwc[c * 3 + 2] + b_dwc[c];
            As[buf][lrA][lcA + j] = ctr + cv;
        }
        float4 wv = *(const float4*)(wrow + k0 + lcB);
        *(float4*)&Bs[buf][lrB][lcB] = wv;
    };

    stage(0, 0);
    __syncthreads();

    int buf = 0;
    for (int k0 = 0; k0 < CH; k0 += 16) {
        if (k0 + 16 < CH) stage(buf ^ 1, k0 + 16);

        const int kb = (lane >> 4) * 2;
        const int ln = lane & 15;
        const int rA = wm * 32;
        const int cB = wn * 32;
#pragma unroll
        for (int kk = 0; kk < 16; kk += 4) {
            v2f a0; a0.x = As[buf][rA + ln][kk + kb];
                    a0.y = As[buf][rA + ln][kk + kb + 1];
            v2f a1; a1.x = As[buf][rA + 16 + ln][kk + kb];
                    a1.y = As[buf][rA + 16 + ln][kk + kb + 1];
            v2f b0; b0.x = Bs[buf][cB + ln][kk + kb];
                    b0.y = Bs[buf][cB + ln][kk + kb + 1];
            v2f b1; b1.x = Bs[buf][cB + 16 + ln][kk + kb];
                    b1.y = Bs[buf][cB + 16 + ln][kk + kb + 1];
            acc00 = wmma4(a0, b0, acc00);
            acc01 = wmma4(a0, b1, acc01);
            acc10 = wmma4(a1, b0, acc10);
            acc11 = wmma4(a1, b1, acc11);
        }
        __syncthreads();
        buf ^= 1;
    }

    const int msel = (lane >> 4) * 8;
    const int ln   = lane & 15;
#pragma unroll
    for (int r = 0; r < 8; ++r) {
        const size_t row0 = (size_t)(blockM + wm * 32 + r + msel);
        const size_t row1 = row0 + 16;
        const int    col  = blockO + wn * 32 + ln;
        out[row0 * CH + col]      = acc00[r] + b_proj[col];
        out[row0 * CH + col + 16] = acc01[r] + b_proj[col + 16];
        out[row1 * CH + col]      = acc10[r] + b_proj[col];
        out[row1 * CH + col + 16] = acc11[r] + b_proj[col + 16];
    }
}

// ---------------------------------------------------------------------------
extern "C" void kernel_launch(void* const* d_in, const int* in_sizes, int n_in,
                              void* d_out, int out_size, void* d_ws, size_t ws_size,
                              hipStream_t stream) {
    (void)in_sizes; (void)n_in; (void)out_size; (void)ws_size;
    const float* x      = (const float*)d_in[0];
    const float* w_qkv  = (const float*)d_in[1];
    const float* w_proj = (const float*)d_in[2];
    const float* b_proj = (const float*)d_in[3];
    const float* w_dwc  = (const float*)d_in[4];
    const float* b_dwc  = (const float*)d_in[5];
    float* out = (float*)d_out;

    float* qkv  = (float*)d_ws;                         // 16384 * 3072 floats
    float* kv   = qkv + (size_t)16384 * C3;             // 64 * 64 * 64 floats
    float* ksum = kv + 64 * 64 * 64;                    // 64 * 64 floats

    fla_qkv_gemm<<<dim3(128, 48), 256, 0, stream>>>(x, w_qkv, qkv);
    fla_featmap <<<65536,          256, 0, stream>>>(qkv);
    fla_ksum    <<<64,             256, 0, stream>>>(qkv, ksum);
    fla_kv      <<<64,             256, 0, stream>>>(qkv, kv);
    fla_apply   <<<dim3(64, 32),   256, 0, stream>>>(qkv, kv, ksum);
    fla_proj    <<<dim3(128, 16),  256, 0, stream>>>(qkv, w_proj, b_proj,
                                                     w_dwc, b_dwc, out);
}